// RiskGNN_22728966930758
// MI455X (gfx1250) — compile-verified
//
#include <hip/hip_runtime.h>
#include <hip/hip_bf16.h>
#include <math.h>

// ---------------------------------------------------------------------------
// RiskGNN: 2-layer GraphSAGE (mean aggr) + linear head + log_softmax
//   N=50000 nodes, E=600000 edges, C=128 features, 8 classes
// Strategy (MI455X / gfx1250, wave32):
//   - scatter-add message aggregation via global_atomic_add_f32 (memory bound)
//   - fused layer GEMM  out = relu( (agg/cnt)@W_l + b + x@W_r )
//     using V_WMMA_F32_16X16X4_F32, one wave per 16x16 output tile,
//     8 waves/block = one 16-row stripe across all 128 output cols
//   - tiny 128->8 head + log_softmax in plain VALU
// ---------------------------------------------------------------------------

#define NN 50000
#define NE 600000
#define CC 128

typedef __attribute__((ext_vector_type(2))) float v2f;
typedef __attribute__((ext_vector_type(8))) float v8f;

// ---- degree counts: cnt[dst] += 1 -----------------------------------------
__global__ void riskgnn_count_kernel(const int* __restrict__ dst,
                                     float* __restrict__ cnt, int E) {
    int e = blockIdx.x * blockDim.x + threadIdx.x;
    if (e < E) atomicAdd(&cnt[dst[e]], 1.0f);
}

// ---- scatter-add: agg[dst] += feat[src] (one wave32 per edge) -------------
__global__ void riskgnn_scatter_kernel(const float* __restrict__ feat,
                                       const int* __restrict__ src,
                                       const int* __restrict__ dst,
                                       float* __restrict__ agg, int E) {
    int t = blockIdx.x * blockDim.x + threadIdx.x;
    int e = t >> 5;
    if (e >= E) return;
    int cg = (t & 31) * 4;                      // 32 lanes x 4 channels = 128
    int s = src[e];
    int d = dst[e];
    const float4 v = *(const float4*)(feat + (size_t)s * CC + cg);
    float* p = agg + (size_t)d * CC + cg;
    atomicAdd(p + 0, v.x);
    atomicAdd(p + 1, v.y);
    atomicAdd(p + 2, v.z);
    atomicAdd(p + 3, v.w);
}

// ---- fused SAGE layer: out = relu((agg/cnt)@Wl + bias + xin@Wr) -----------
// One wave computes one 16x16 tile of the [N,128] output.
// blockDim=256 (8 waves): block b covers rows [16b,16b+16), waves cover the
// 8 column tiles. WMMA fragment layouts per CDNA5 ISA 7.12.2:
//   A (16x4 f32): lanes 0-15 -> K=koff..koff+1 with koff=0, lanes 16-31 koff=2
//   B (4x16 f32): lane holds B[k+koff][col], B[k+koff+1][col], col=lane&15
//   C/D: VGPR i -> M = i + (lane<16 ? 0 : 8), N = lane&15
__global__ __launch_bounds__(256)
void riskgnn_layer_kernel(const float* __restrict__ xin,   // [N,128] self
                          const float* __restrict__ agg,   // [N,128] msg sums
                          const float* __restrict__ cnt,   // [N]
                          const float* __restrict__ Wl,    // [128,128] row-major
                          const float* __restrict__ Wr,    // [128,128]
                          const float* __restrict__ bias,  // [128]
                          float* __restrict__ out)         // [N,128]
{
    const int lane = threadIdx.x & 31;
    const int wave = threadIdx.x >> 5;        // 0..7 -> column tile
    const int row0 = blockIdx.x * 16;
    const int m    = lane & 15;               // row within tile
    const int koff = (lane < 16) ? 0 : 2;     // K sub-offset of this half-wave
    const int col  = wave * 16 + (lane & 15); // output column for B/C/D frags

    const int row   = row0 + m;
    const float c   = cnt[row];
    const float inv = 1.0f / fmaxf(c, 1.0f);  // mean = sum / max(cnt,1)

    const float* arow  = agg + (size_t)row * CC + koff;
    const float* xrow  = xin + (size_t)row * CC + koff;
    const float* blcol = Wl + col;            // walk K with stride 128
    const float* brcol = Wr + col;

    v8f acc = {};

    // (agg/cnt) @ Wl  : K = 0..127 in steps of 4
#pragma unroll
    for (int k = 0; k < CC; k += 4) {
        v2f a, b;
        a.x = arow[k]     * inv;
        a.y = arow[k + 1] * inv;
        b.x = blcol[(size_t)(k + koff)     * CC];
        b.y = blcol[(size_t)(k + koff + 1) * CC];
        acc = __builtin_amdgcn_wmma_f32_16x16x4_f32(
            false, a, false, b, (short)0, acc, false, false);
    }
    // x @ Wr : accumulate into same tile
#pragma unroll
    for (int k = 0; k < CC; k += 4) {
        v2f a, b;
        a.x = xrow[k];
        a.y = xrow[k + 1];
        b.x = brcol[(size_t)(k + koff)     * CC];
        b.y = brcol[(size_t)(k + koff + 1) * CC];
        acc = __builtin_amdgcn_wmma_f32_16x16x4_f32(
            false, a, false, b, (short)0, acc, false, false);
    }

    const float bb = bias[col];
#pragma unroll
    for (int i = 0; i < 8; ++i) {
        float v = fmaxf(acc[i] + bb, 0.0f);   // relu
        int orow = row0 + i + ((lane < 16) ? 0 : 8);
        out[(size_t)orow * CC + col] = v;
    }
}

// ---- head: logits = h@Wlin + blin ; out = log_softmax(logits) -------------
__global__ void riskgnn_head_kernel(const float* __restrict__ h,    // [N,128]
                                    const float* __restrict__ Wlin, // [128,8]
                                    const float* __restrict__ blin, // [8]
                                    float* __restrict__ out, int N) {
    int i = blockIdx.x * blockDim.x + threadIdx.x;
    if (i >= N) return;
    float acc[8];
#pragma unroll
    for (int j = 0; j < 8; ++j) acc[j] = blin[j];
    const float* hr = h + (size_t)i * CC;
    for (int k = 0; k < CC; ++k) {
        float hv = hr[k];
        const float* w = Wlin + k * 8;        // uniform across lanes -> cached
#pragma unroll
        for (int j = 0; j < 8; ++j) acc[j] = fmaf(hv, w[j], acc[j]);
    }
    float mx = acc[0];
#pragma unroll
    for (int j = 1; j < 8; ++j) mx = fmaxf(mx, acc[j]);
    float s = 0.0f;
#pragma unroll
    for (int j = 0; j < 8; ++j) s += expf(acc[j] - mx);
    float lse = mx + logf(s);
    float* o = out + (size_t)i * 8;
#pragma unroll
    for (int j = 0; j < 8; ++j) o[j] = acc[j] - lse;
}

extern "C" void kernel_launch(void* const* d_in, const int* in_sizes, int n_in,
                              void* d_out, int out_size, void* d_ws, size_t ws_size,
                              hipStream_t stream) {
    const float* x    = (const float*)d_in[0];
    const int*   ei   = (const int*)d_in[1];   // [2, E]
    const float* W1l  = (const float*)d_in[2];
    const float* b1   = (const float*)d_in[3];
    const float* W1r  = (const float*)d_in[4];
    const float* W2l  = (const float*)d_in[5];
    const float* b2   = (const float*)d_in[6];
    const float* W2r  = (const float*)d_in[7];
    const float* Wlin = (const float*)d_in[8];
    const float* blin = (const float*)d_in[9];
    float* out = (float*)d_out;

    const int N = NN, E = NE;
    const int* src = ei;
    const int* dst = ei + E;

    // workspace layout: agg[N*128] | cnt[N] | h1[N*128] | h2[N*128]  (~77 MB)
    float* agg = (float*)d_ws;
    float* cnt = agg + (size_t)N * CC;
    float* h1  = cnt + N;
    float* h2  = h1 + (size_t)N * CC;

    // zero agg + cnt (capture-safe)
    hipMemsetAsync(agg, 0, ((size_t)N * CC + N) * sizeof(float), stream);

    riskgnn_count_kernel<<<(E + 255) / 256, 256, 0, stream>>>(dst, cnt, E);

    // layer 1
    riskgnn_scatter_kernel<<<((size_t)E * 32 + 255) / 256, 256, 0, stream>>>(
        x, src, dst, agg, E);
    riskgnn_layer_kernel<<<N / 16, 256, 0, stream>>>(x, agg, cnt, W1l, W1r, b1, h1);

    // layer 2 (reuse cnt; re-zero agg)
    hipMemsetAsync(agg, 0, (size_t)N * CC * sizeof(float), stream);
    riskgnn_scatter_kernel<<<((size_t)E * 32 + 255) / 256, 256, 0, stream>>>(
        h1, src, dst, agg, E);
    riskgnn_layer_kernel<<<N / 16, 256, 0, stream>>>(h1, agg, cnt, W2l, W2r, b2, h2);

    // head + log_softmax
    riskgnn_head_kernel<<<(N + 255) / 256, 256, 0, stream>>>(h2, Wlin, blin, out, N);
}